// LinearAssignmentLossCE_49306224558372
// MI455X (gfx1250) — compile-verified
//
#include <hip/hip_runtime.h>
#include <math.h>

#define NSEG  32
#define CHUNK 512
#define TPB   256

// ---------------- CDNA5 async global->LDS path (probe with fallback) --------
#if __has_builtin(__builtin_amdgcn_global_load_async_to_lds_b32)
#define HAVE_ASYNC_LDS 1
#warning "CDNA5 probe: __builtin_amdgcn_global_load_async_to_lds_b32 AVAILABLE"
#else
#define HAVE_ASYNC_LDS 0
#warning "CDNA5 probe: async-to-LDS builtin NOT available; sync LDS fallback"
#endif

typedef __attribute__((address_space(1))) int GI;  // global int*
typedef __attribute__((address_space(3))) int LI;  // LDS int*

__device__ __forceinline__ void async_cp_b32(void* lds, const void* g) {
#if HAVE_ASYNC_LDS
  void* gnc = (void*)g;  // builtin takes non-const global int*
  __builtin_amdgcn_global_load_async_to_lds_b32((GI*)gnc, (LI*)lds, 0, 0);
#else
  *(volatile int*)lds = *(const int*)g;   // synchronous fallback
#endif
}

template <int N>
__device__ __forceinline__ void wait_async() {
#if __has_builtin(__builtin_amdgcn_s_wait_asynccnt)
  __builtin_amdgcn_s_wait_asynccnt(N);
#else
  asm volatile("s_wait_asynccnt %0" ::"i"(N) : "memory");
#endif
}

// ---------------- Kernel 1: per-(target, segment) online logsumexp ----------
// LDS layout: interleaved 16B records {src, dst, score_bits, pad} so the hot
// loop does ONE ds_load_b128 per edge instead of three ds_load_b32.
__global__ void __launch_bounds__(TPB)
lal_partial_kernel(const int* __restrict__ edge_index,   // [2,E]
                   const float* __restrict__ score,      // [E]
                   const int* __restrict__ target_edges, // [2,T]
                   int E, int T,
                   float* __restrict__ ws_m, float* __restrict__ ws_s,
                   int* __restrict__ ws_k, int* __restrict__ ws_idx) {
  __shared__ int4 lpack[2][CHUNK];   // 16 KB total

  const int tgt = blockIdx.x * TPB + threadIdx.x;  // one target per lane
  const int seg = blockIdx.y;
  const int segLen = (E + NSEG - 1) / NSEG;
  const int segBeg = seg * segLen;
  const int segEnd = min(E, segBeg + segLen);

  const int* src = edge_index;
  const int* dst = edge_index + E;

  int tsrc = 0, tdst = 0;
  if (tgt < T) {
    tsrc = target_edges[tgt];
    tdst = target_edges[T + tgt];
  }

  float m = -INFINITY, ssum = 0.0f;
  int K = 0, minidx = 0x7FFFFFFF;

  const int nchunks = (segEnd - segBeg + CHUNK - 1) / CHUNK;

  // Prefetch chunk 0: 6 scattered async b32 per thread into packed records.
  for (int j = (int)threadIdx.x; j < CHUNK; j += TPB) {
    const int g = min(segBeg + j, E - 1);
    async_cp_b32(&lpack[0][j].x, &src[g]);
    async_cp_b32(&lpack[0][j].y, &dst[g]);
    async_cp_b32(&lpack[0][j].z, &score[g]);
  }

  for (int c = 0; c < nchunks; ++c) {
    const int buf = c & 1;
    if (c + 1 < nchunks) {
      // Issue next chunk into the other buffer, then wait until only those
      // 6 per-thread async ops remain outstanding (in-order completion).
      const int base = segBeg + (c + 1) * CHUNK;
      for (int j = (int)threadIdx.x; j < CHUNK; j += TPB) {
        const int g = min(base + j, E - 1);
        async_cp_b32(&lpack[buf ^ 1][j].x, &src[g]);
        async_cp_b32(&lpack[buf ^ 1][j].y, &dst[g]);
        async_cp_b32(&lpack[buf ^ 1][j].z, &score[g]);
      }
      wait_async<6>();
    } else {
      wait_async<0>();
    }
    __syncthreads();

    const int base = segBeg + c * CHUNK;
    const int cnt  = min(CHUNK, segEnd - base);
    if (tgt < T) {
      for (int i = 0; i < cnt; ++i) {
        const int4 e = lpack[buf][i];   // single ds_load_b128, LDS broadcast
        const bool ms = (e.x == tsrc);
        const bool md = (e.y == tdst);
        if (ms | md) {                  // ~0.02% taken: cheap divergence
          const float sc = __int_as_float(e.z);
          ++K;
          if (sc > m) {                 // online logsumexp update
            ssum = ssum * __expf(m - sc) + 1.0f;
            m = sc;
          } else {
            ssum += __expf(sc - m);
          }
          if (ms & md) minidx = min(minidx, base + i);
        }
      }
    }
    __syncthreads();  // all lanes done reading buf before it is overwritten
  }

  if (tgt < T) {
    const int o = tgt * NSEG + seg;
    ws_m[o] = m; ws_s[o] = ssum; ws_k[o] = K; ws_idx[o] = minidx;
  }
}

// ---------------- Kernel 2: deterministic merge + reduction -----------------
__global__ void __launch_bounds__(1024)
lal_reduce_kernel(const float* __restrict__ score,
                  const float* __restrict__ ws_m, const float* __restrict__ ws_s,
                  const int* __restrict__ ws_k, const int* __restrict__ ws_idx,
                  int T, float* __restrict__ out) {
  __shared__ float red[1024];
  float acc = 0.0f;

  for (int t = (int)threadIdx.x; t < T; t += (int)blockDim.x) {
    float M = -INFINITY;
    int K = 0, idx = 0x7FFFFFFF;
    for (int s = 0; s < NSEG; ++s) {
      const int o = t * NSEG + s;
      M = fmaxf(M, ws_m[o]);
      K += ws_k[o];
      idx = min(idx, ws_idx[o]);
    }
    if (K > 0) {
      float S = 0.0f;
      for (int s = 0; s < NSEG; ++s) {
        const int o = t * NSEG + s;
        const float sv = ws_s[o];
        if (sv > 0.0f) S += sv * __expf(ws_m[o] - M);
      }
      const float lse = M + __logf(S);
      acc += (lse - score[idx]) / (float)K;
    }
  }

  red[threadIdx.x] = acc;
  __syncthreads();
  for (int off = (int)blockDim.x >> 1; off > 0; off >>= 1) {
    if ((int)threadIdx.x < off) red[threadIdx.x] += red[threadIdx.x + off];
    __syncthreads();
  }
  if (threadIdx.x == 0) out[0] = red[0] / (float)T;
}

// ---------------- Host-side launcher ----------------------------------------
extern "C" void kernel_launch(void* const* d_in, const int* in_sizes, int n_in,
                              void* d_out, int out_size, void* d_ws, size_t ws_size,
                              hipStream_t stream) {
  const int*   edge_index   = (const int*)d_in[0];    // [2,E] int32
  const float* score        = (const float*)d_in[1];  // [E]   f32
  const int*   target_edges = (const int*)d_in[2];    // [2,T] int32
  // d_in[3] = num_nodes (unused)

  const int E = in_sizes[1];
  const int T = in_sizes[2] / 2;

  // Workspace layout: m | s | k | idx, each T*NSEG entries (2 MB total here).
  const size_t n = (size_t)T * NSEG;
  float* ws_m   = (float*)d_ws;
  float* ws_s   = ws_m + n;
  int*   ws_k   = (int*)(ws_s + n);
  int*   ws_idx = ws_k + n;

  dim3 grid1((T + TPB - 1) / TPB, NSEG);
  lal_partial_kernel<<<grid1, TPB, 0, stream>>>(
      edge_index, score, target_edges, E, T, ws_m, ws_s, ws_k, ws_idx);

  lal_reduce_kernel<<<1, 1024, 0, stream>>>(
      score, ws_m, ws_s, ws_k, ws_idx, T, (float*)d_out);
}